// LocallyConnectedLayer_17403207483437
// MI455X (gfx1250) — compile-verified
//
#include <hip/hip_runtime.h>

typedef __attribute__((ext_vector_type(2))) float v2f;
typedef __attribute__((ext_vector_type(8))) float v8f;
typedef __attribute__((ext_vector_type(2))) int   v2i;

#if defined(__has_builtin)
#  if __has_builtin(__builtin_amdgcn_global_load_async_to_lds_b64) && \
      __has_builtin(__builtin_amdgcn_global_load_async_to_lds_b32)
#    define HAVE_ASYNC_LDS 1
#  endif
#endif
#ifndef HAVE_ASYNC_LDS
#  define HAVE_ASYNC_LDS 0
#endif

#define GA_V2I(p) ((__attribute__((address_space(1))) v2i*)(p))
#define LA_V2I(p) ((__attribute__((address_space(3))) v2i*)(p))
#define GA_I32(p) ((__attribute__((address_space(1))) int*)(p))
#define LA_I32(p) ((__attribute__((address_space(3))) int*)(p))

#if HAVE_ASYNC_LDS
#  if __has_builtin(__builtin_amdgcn_s_wait_asynccnt)
#    define WAIT_ASYNC(n) __builtin_amdgcn_s_wait_asynccnt(n)
#  else
#    define WAIT_ASYNC(n) asm volatile("s_wait_asynccnt %0" :: "i"(n) : "memory")
#  endif
#else
#  define WAIT_ASYNC(n) do {} while (0)
#endif

#define BB     128
#define CINC   64
#define HH     32
#define WWD    32
#define COUTC  64
#define KF     3
#define KTOT   (CINC * KF * KF)   // 576
#define CCH    8                  // channels per K-chunk
#define NCHUNK (CINC / CCH)       // 8
#define KC     (CCH * KF * KF)    // 72
#define NG     (KC / 4)           // 18 wmma K-groups per chunk

#define ASZ (8 * NG * 64)         // 9216 floats per A buffer (36 KB)
#define BSZ (NG * 4 * 64)         // 4608 floats per B buffer (18 KB)
#define ASYNC_PER_CHUNK 45        // 9 b64 (B) + 36 b32 (A) per thread/wave

// compile-time helpers for the A-item constants (kl = k within chunk, 0..71)
__device__ constexpr int a_goff(int kl) {            // global float offset inside chunk
    return (kl / 9) * (HH * WWD) + ((kl % 9) / 3) * WWD + (kl % 3);
}
__device__ constexpr int a_lds(int kl) {             // swizzled LDS float offset inside wave strip
    return (kl >> 2) * 64 + ((kl & 3) >> 1) * 32 + (kl & 1);
}
__device__ constexpr int a_p(int kl) { return (kl % 9) / 3; }
__device__ constexpr int a_q(int kl) { return kl % 3; }

__global__ __launch_bounds__(256)
void lcl_wmma_f32_kernel(const float* __restrict__ x,
                         const float* __restrict__ weight,
                         const float* __restrict__ bias,
                         float* __restrict__ out)
{
    __shared__ float ldsA[2 * ASZ];      // 72 KB, WMMA-swizzled patches (double buffered)
    __shared__ float ldsB[2 * BSZ];      // 36 KB, WMMA-swizzled weights (double buffered)
    __shared__ float ldsDummy[32];       // sink for OOB async lanes

    const int w   = blockIdx.x;
    const int h   = blockIdx.y;
    const int tid = threadIdx.x;
    const int wv  = tid >> 5;
    const int ln  = tid & 31;

    const float* wpos = weight + (size_t)(h * WWD + w) * COUTC * KTOT;

    // ---- per-thread copy-item bases (division-free bit decompositions) ----
    // B items: n = tid[6:1], half = tid[0], g = 2*jj + tid[7]
    const int nB     = (tid >> 1) & 63;
    const int halfB  = tid & 1;
    const int gpB    = tid >> 7;
    const int bgBase = nB * KTOT + gpB * 4 + halfB * 2;                    // + ck*72 + jj*8
    const int blBase = (gpB * 4 + (nB >> 4)) * 64 + ((nB & 15) + halfB * 16) * 2;  // + jj*512
    // A items: b = tid[6:0], kl = 2*jj + tid[7]
    const int bA     = tid & 127;
    const int sel    = tid >> 7;
    const float* xb  = x + (size_t)bA * (CINC * HH * WWD);
    const int alBase = ((tid >> 4) & 7) * (NG * 64) + (tid & 15) * 2;      // wave strip + m
    float* dummy     = ldsDummy + ln;

    v8f acc[4];
    #pragma unroll
    for (int t = 0; t < 4; ++t)
        #pragma unroll
        for (int e = 0; e < 8; ++e) acc[t][e] = 0.0f;

    // Pre-zero both A buffers: zero-padded slots are never overwritten by copies.
    #pragma unroll
    for (int i = tid; i < (2 * ASZ) / 4; i += 256)
        reinterpret_cast<float4*>(ldsA)[i] = make_float4(0.f, 0.f, 0.f, 0.f);
    __syncthreads();

    // ---- rotated software pipeline: issue(ck) overlaps compute(ck-1) ----
    for (int ck = 0; ck <= NCHUNK; ++ck) {
        if (ck < NCHUNK) {
            float* bufA = ldsA + (ck & 1) * ASZ;
            float* bufB = ldsB + (ck & 1) * BSZ;
            const int uoff = ck * (CCH * HH * WWD) + (h - 1) * WWD + (w - 1); // chunk-uniform

            // B (weight): 9 async b64 swizzled copies per thread
            #pragma unroll
            for (int jj = 0; jj < 9; ++jj) {
                const float* gsrc = wpos + (bgBase + ck * KC + jj * 8);
                float*       ldst = bufB + (blBase + jj * 512);
#if HAVE_ASYNC_LDS
                __builtin_amdgcn_global_load_async_to_lds_b64(GA_V2I(gsrc), LA_V2I(ldst), 0, 0);
#else
                *reinterpret_cast<float2*>(ldst) = *reinterpret_cast<const float2*>(gsrc);
#endif
            }

            // A (patches): 36 async b32 scatter copies per thread; constants fold per jj
            #pragma unroll
            for (int jj = 0; jj < 36; ++jj) {
                const int kl0 = jj * 2, kl1 = jj * 2 + 1;
                const int ok0 = ((unsigned)(h - 1 + a_p(kl0)) < (unsigned)HH) &&
                                ((unsigned)(w - 1 + a_q(kl0)) < (unsigned)WWD);
                const int ok1 = ((unsigned)(h - 1 + a_p(kl1)) < (unsigned)HH) &&
                                ((unsigned)(w - 1 + a_q(kl1)) < (unsigned)WWD);
                const int okm  = ok0 + sel * (ok1 - ok0);
                const int goff = a_goff(kl0) + sel * (a_goff(kl1) - a_goff(kl0));
                const int loff = a_lds(kl0)  + sel * (a_lds(kl1)  - a_lds(kl0));
                const float* gsrc = xb + (okm ? (uoff + goff) : 0);   // clamped when OOB
                float*       ldst = okm ? (bufA + alBase + loff) : dummy;
#if HAVE_ASYNC_LDS
                __builtin_amdgcn_global_load_async_to_lds_b32(GA_I32(gsrc), LA_I32(ldst), 0, 0);
#else
                if (okm) *ldst = *gsrc;
#endif
            }
        }

        if (ck == 0) continue;

        if (ck < NCHUNK) { WAIT_ASYNC(ASYNC_PER_CHUNK); }  // chunk ck-1 copies have landed
        else             { WAIT_ASYNC(0); }
        __syncthreads();

        const int cc = ck - 1;
        const float* curA = ldsA + (cc & 1) * ASZ;
        const float* curB = ldsB + (cc & 1) * BSZ;

        #pragma unroll 3
        for (int g = 0; g < NG; ++g) {
            const v2f a = *reinterpret_cast<const v2f*>(&curA[(wv * NG + g) * 64 + ln * 2]);
            #pragma unroll
            for (int t = 0; t < 4; ++t) {
                const v2f bop = *reinterpret_cast<const v2f*>(&curB[(g * 4 + t) * 64 + ln * 2]);
                acc[t] = __builtin_amdgcn_wmma_f32_16x16x4_f32(
                    false, a, false, bop, (short)0, acc[t], false, false);
            }
        }

        __syncthreads();   // buffer (ck-1)&1 may be refilled next iteration
    }

    // ---- epilogue: add bias[d,h,w], store out[b,d,h,w]
    const int hw = h * WWD + w;
    #pragma unroll
    for (int t = 0; t < 4; ++t) {
        const int n    = t * 16 + (ln & 15);
        const float bv = bias[n * (HH * WWD) + hw];
        const int mbase = (ln >> 4) * 8;
        #pragma unroll
        for (int r = 0; r < 8; ++r) {
            const int b = wv * 16 + mbase + r;
            out[((size_t)b * COUTC + n) * (HH * WWD) + hw] = acc[t][r] + bv;
        }
    }
}

extern "C" void kernel_launch(void* const* d_in, const int* in_sizes, int n_in,
                              void* d_out, int out_size, void* d_ws, size_t ws_size,
                              hipStream_t stream) {
    const float* x      = (const float*)d_in[0];
    const float* weight = (const float*)d_in[1];
    const float* bias   = (const float*)d_in[2];
    float* out          = (float*)d_out;

    dim3 grid(WWD, HH);   // one workgroup per output position (w, h)
    lcl_wmma_f32_kernel<<<grid, 256, 0, stream>>>(x, weight, bias, out);
}